// ASPPPooling_38268158607596
// MI455X (gfx1250) — compile-verified
//
#include <hip/hip_runtime.h>

typedef __attribute__((ext_vector_type(2))) float v2f;
typedef __attribute__((ext_vector_type(4))) float v4f;
typedef __attribute__((ext_vector_type(8))) float v8f;

#define BN_EPS 1e-5f
constexpr int Bsz  = 16;
constexpr int CIN  = 2048;
constexpr int HW   = 4096;   // 64*64
constexpr int COUT = 256;

// ---------------------------------------------------------------------------
// Stage 1: global average pool.  One wave32 per (b, cin) channel.
// 4096 floats/channel -> 32 lanes * 32 float4 NT loads, then shuffle reduce.
// This stage moves 536 MB and is the bandwidth roofline of the whole problem.
// ---------------------------------------------------------------------------
__global__ void pool_kernel(const float* __restrict__ x,
                            float* __restrict__ pooled) {
  const int wave = threadIdx.x >> 5;              // 8 waves / block
  const int lane = threadIdx.x & 31;
  const int ch   = blockIdx.x * 8 + wave;         // [0, 16*2048)
  const v4f* src = reinterpret_cast<const v4f*>(x + (size_t)ch * HW);

  v4f acc = {};
  #pragma unroll
  for (int i = 0; i < 32; ++i) {
    v4f v = __builtin_nontemporal_load(&src[i * 32 + lane]);
    acc += v;                                     // ext-vector elementwise
  }
  float s = (acc.x + acc.y) + (acc.z + acc.w);
  #pragma unroll
  for (int off = 16; off > 0; off >>= 1)
    s += __shfl_xor(s, off, 32);
  if (lane == 0)
    pooled[ch] = s * (1.0f / (float)HW);
}

// ---------------------------------------------------------------------------
// Stage 2: Y[16,256] = pooled[16,2048] @ W[256,2048]^T, fused BN(eval)+ReLU.
// Full-precision fp32 matrix pipe: V_WMMA_F32_16X16X4_F32, K stepped by 4.
// One block of 512 threads = 16 waves; wave w owns N-tile [16w, 16w+16).
// A lane layout (ISA 7.12.2, 32-bit A 16x4): lanes 0-15 K={k,k+1}, M=lane;
// lanes 16-31 K={k+2,k+3}, M=lane-16. B mirrors over N. C/D: VGPR v holds
// M = v + 8*(lane>=16), N = lane&15.
// ---------------------------------------------------------------------------
__global__ void gemm_bn_relu_kernel(const float* __restrict__ pooled, // [16][2048]
                                    const float* __restrict__ Wm,     // [256][2048]
                                    const float* __restrict__ gamma,
                                    const float* __restrict__ beta,
                                    const float* __restrict__ mean,
                                    const float* __restrict__ var,
                                    float* __restrict__ y) {          // [16][256]
  const int wave = threadIdx.x >> 5;
  const int lane = threadIdx.x & 31;
  const int half = lane >> 4;          // 0: K pair {0,1}, 1: K pair {2,3}
  const int l16  = lane & 15;
  const int m    = l16;                // A row held by this lane
  const int n    = wave * 16 + l16;    // B column held by this lane

  const float* aRow = pooled + (size_t)m * CIN + half * 2;
  const float* bRow = Wm     + (size_t)n * CIN + half * 2;

  v8f acc = {};
  #pragma unroll 4
  for (int k = 0; k < CIN; k += 4) {
    v2f a = *reinterpret_cast<const v2f*>(aRow + k);   // global_load_b64
    v2f b = *reinterpret_cast<const v2f*>(bRow + k);
    acc = __builtin_amdgcn_wmma_f32_16x16x4_f32(
        /*neg_a=*/false, a, /*neg_b=*/false, b,
        /*c_mod=*/(short)0, acc, /*reuse_a=*/false, /*reuse_b=*/false);
  }

  // BatchNorm (eval) folded to scale/shift, then ReLU. Each lane has fixed n.
  const float sc = gamma[n] * rsqrtf(var[n] + BN_EPS);
  const float sh = beta[n] - mean[n] * sc;
  #pragma unroll
  for (int v = 0; v < 8; ++v) {
    const int mm = v + half * 8;
    float r = fmaxf(acc[v] * sc + sh, 0.0f);
    y[mm * COUT + n] = r;
  }
}

// ---------------------------------------------------------------------------
// Stage 3: broadcast y[b,c] over 64x64 spatial (bilinear-from-1x1 == copy).
// 64 MB write-once -> non-temporal b128 stores. One block per (b,c).
// ---------------------------------------------------------------------------
__global__ void broadcast_kernel(const float* __restrict__ y,
                                 float* __restrict__ out) {
  const int bc   = blockIdx.x;                    // [0, 16*256)
  const float s  = y[bc];
  v4f v = {s, s, s, s};
  v4f* dst = reinterpret_cast<v4f*>(out + (size_t)bc * HW);
  #pragma unroll
  for (int j = 0; j < 4; ++j)
    __builtin_nontemporal_store(v, &dst[j * 256 + threadIdx.x]);
}

extern "C" void kernel_launch(void* const* d_in, const int* in_sizes, int n_in,
                              void* d_out, int out_size, void* d_ws, size_t ws_size,
                              hipStream_t stream) {
  const float* x     = (const float*)d_in[0];   // [16,2048,64,64]
  const float* Wm    = (const float*)d_in[1];   // [256,2048]
  const float* gamma = (const float*)d_in[2];   // [256]
  const float* beta  = (const float*)d_in[3];   // [256]
  const float* mean  = (const float*)d_in[4];   // [256]
  const float* var   = (const float*)d_in[5];   // [256]
  float* out = (float*)d_out;                   // [16,256,64,64]

  float* pooled = (float*)d_ws;                 // 16*2048 floats = 128 KB
  float* y      = pooled + Bsz * CIN;           // 16*256 floats

  // Stage 1: 32768 channels, 8 waves (channels) per 256-thread block.
  pool_kernel<<<(Bsz * CIN) / 8, 256, 0, stream>>>(x, pooled);

  // Stage 2: single block, 16 waves, WMMA fp32 GEMM + BN + ReLU.
  gemm_bn_relu_kernel<<<1, 512, 0, stream>>>(pooled, Wm, gamma, beta, mean, var, y);

  // Stage 3: 4096 (b,c) blocks broadcast into 64x64 tiles.
  broadcast_kernel<<<Bsz * COUT, 256, 0, stream>>>(y, out);
}